// InferenceTransform_8014408974602
// MI455X (gfx1250) — compile-verified
//
#include <hip/hip_runtime.h>
#include <hip/hip_bf16.h>
#include <math.h>
#include <stdint.h>

#define CLS_THRESH 0.05f
#define FXR 0.1f
#define FYR 0.1f
#define FWR 0.2f
#define FHR 0.2f

#define NBUF 4   // LDS row buffers per wave
#define PD   3   // prefetch distance (rows in flight beyond current)

// ---------------------------------------------------------------------------
// Fused epilogue: box decode + clip + mask + all output writes (lane 0 only).
// Output layout (floats): [boxes B*N*4][scores B*N][cls_idx B*N][mask B*N]
// ---------------------------------------------------------------------------
__device__ __forceinline__ void write_row_outputs(
    float* __restrict__ out, int nRows, int row, int n,
    float score, int bidx,
    const float4* __restrict__ reg4, const float4* __restrict__ anc4,
    float imgW, float imgH)
{
    const float m = (score > CLS_THRESH) ? 1.0f : 0.0f;

    const float4 a = anc4[n];
    const float4 r = reg4[row];

    const float w  = a.z - a.x;
    const float h  = a.w - a.y;
    const float cx = a.x + 0.5f * w;
    const float cy = a.y + 0.5f * h;

    const float pcx = cx + (r.x * FXR) * w;
    const float pcy = cy + (r.y * FYR) * h;
    const float pw  = expf(r.z * FWR) * w;
    const float ph  = expf(r.w * FHR) * h;

    float4 box;
    box.x = fminf(fmaxf(pcx - 0.5f * pw, 0.0f), imgW) * m;
    box.y = fminf(fmaxf(pcy - 0.5f * ph, 0.0f), imgH) * m;
    box.z = fminf(fmaxf(pcx + 0.5f * pw, 0.0f), imgW) * m;
    box.w = fminf(fmaxf(pcy + 0.5f * ph, 0.0f), imgH) * m;

    ((float4*)out)[row] = box;                       // boxes_m
    float* scores = out + (size_t)4 * nRows;
    scores[row] = score * m;                         // scores_m
    float* clsi = scores + nRows;
    clsi[row] = (float)bidx;                         // cls_idx
    float* mask = clsi + nRows;
    mask[row] = m;                                   // mask
}

// ---------------------------------------------------------------------------
// Async global -> LDS issue of one 2000-byte class row (C == 500).
// 4 x global_load_async_to_lds_b128 (NT: 786MB streamed once, don't thrash
// the 192MB L2). Each lane moves 16B -> 512B per chunk per wave.
// Lanes addressing float4 index >= 125 clamp to row base (slots never read).
// ---------------------------------------------------------------------------
__device__ __forceinline__ void issue_row_async(
    const float* __restrict__ cls, int row, float4* ldsBuf, int lane)
{
    const unsigned long long rowBase =
        (unsigned long long)(uintptr_t)cls + (unsigned long long)row * 2000ull;
#pragma unroll
    for (int c = 0; c < 4; ++c) {
        const int i4 = c * 32 + lane;
        const unsigned ldsAddr = (unsigned)(uintptr_t)(ldsBuf + i4);
        const unsigned long long g =
            rowBase + (unsigned long long)((i4 < 125 ? i4 : 0) * 16);
        asm volatile("global_load_async_to_lds_b128 %0, %1, off th:TH_LOAD_NT"
                     :: "v"(ldsAddr), "v"(g)
                     : "memory");
    }
}

// Exact retire fence for the oldest in-flight row. `remBehind` = rows still
// in flight NOT counting the one we are about to consume (wave-uniform).
__device__ __forceinline__ void wait_oldest_row(int remBehind)
{
    if (remBehind >= 3)      asm volatile("s_wait_asynccnt 12" ::: "memory");
    else if (remBehind == 2) asm volatile("s_wait_asynccnt 8"  ::: "memory");
    else if (remBehind == 1) asm volatile("s_wait_asynccnt 4"  ::: "memory");
    else                     asm volatile("s_wait_asynccnt 0"  ::: "memory");
}

// ---------------------------------------------------------------------------
// Fast path (C == 500): one wave per row, quad-buffered async LDS staging,
// prefetch distance 3. 8 waves/block, 4 x 2048B buffers/wave = 64KB/block.
// ---------------------------------------------------------------------------
__global__ __launch_bounds__(256)
void argmax_box_async_kernel(const float* __restrict__ cls,
                             const float* __restrict__ reg,
                             const float* __restrict__ anc,
                             float* __restrict__ out,
                             int nRows, int N, float imgW, float imgH)
{
    __shared__ float4 smem[8 * NBUF * 128];          // 64 KB

    const int lane   = threadIdx.x & 31;
    const int lw     = threadIdx.x >> 5;             // wave in block (0..7)
    const int gw     = blockIdx.x * (blockDim.x >> 5) + lw;
    const int stride = gridDim.x * (blockDim.x >> 5);

    float4* bufs = &smem[lw * NBUF * 128];

    const float4* reg4 = (const float4*)reg;
    const float4* anc4 = (const float4*)anc;

    // Prologue: put up to PD rows in flight.
    int inflight = 0;
#pragma unroll
    for (int j = 0; j < PD; ++j) {
        const int r = gw + j * stride;
        if (r < nRows) {
            issue_row_async(cls, r, bufs + (j & (NBUF - 1)) * 128, lane);
            ++inflight;
        }
    }

    int j = 0;
    for (int row = gw; row < nRows; row += stride, ++j) {
        const int rn = row + PD * stride;            // wave-uniform
        if (rn < nRows) {
            issue_row_async(cls, rn, bufs + ((j + PD) & (NBUF - 1)) * 128, lane);
            ++inflight;
        }
        wait_oldest_row(inflight - 1);               // retire row `row`

        const float4* b = bufs + (j & (NBUF - 1)) * 128;

        float best = -INFINITY;
        int   bidx = 0;
#pragma unroll
        for (int c = 0; c < 4; ++c) {
            const int i4 = c * 32 + lane;
            if (i4 < 125) {
                const float4 v = b[i4];              // ds_load_b128
                const int base = i4 * 4;
                if (v.x > best) { best = v.x; bidx = base;     }
                if (v.y > best) { best = v.y; bidx = base + 1; }
                if (v.z > best) { best = v.z; bidx = base + 2; }
                if (v.w > best) { best = v.w; bidx = base + 3; }
            }
        }
        // wave32 butterfly reduce; ties -> smallest class index (argmax-first)
#pragma unroll
        for (int mlane = 16; mlane >= 1; mlane >>= 1) {
            const float ov = __shfl_xor(best, mlane, 32);
            const int   oi = __shfl_xor(bidx, mlane, 32);
            if (ov > best || (ov == best && oi < bidx)) { best = ov; bidx = oi; }
        }

        if (lane == 0) {
            write_row_outputs(out, nRows, row, row % N, best, bidx,
                              reg4, anc4, imgW, imgH);
        }
        --inflight;
    }
}

// ---------------------------------------------------------------------------
// Generic fallback (any C): one wave per row, coalesced scalar loads.
// ---------------------------------------------------------------------------
__global__ __launch_bounds__(256)
void argmax_box_generic_kernel(const float* __restrict__ cls,
                               const float* __restrict__ reg,
                               const float* __restrict__ anc,
                               float* __restrict__ out,
                               int nRows, int N, int C,
                               float imgW, float imgH)
{
    const int lane   = threadIdx.x & 31;
    const int gw     = blockIdx.x * (blockDim.x >> 5) + (threadIdx.x >> 5);
    const int stride = gridDim.x * (blockDim.x >> 5);

    const float4* reg4 = (const float4*)reg;
    const float4* anc4 = (const float4*)anc;

    for (int row = gw; row < nRows; row += stride) {
        const float* p = cls + (size_t)row * (size_t)C;
        float best = -INFINITY;
        int   bidx = 0;
        for (int j = lane; j < C; j += 32) {
            const float v = p[j];
            if (v > best) { best = v; bidx = j; }
        }
#pragma unroll
        for (int mlane = 16; mlane >= 1; mlane >>= 1) {
            const float ov = __shfl_xor(best, mlane, 32);
            const int   oi = __shfl_xor(bidx, mlane, 32);
            if (ov > best || (ov == best && oi < bidx)) { best = ov; bidx = oi; }
        }
        if (lane == 0) {
            write_row_outputs(out, nRows, row, row % N, best, bidx,
                              reg4, anc4, imgW, imgH);
        }
    }
}

// ---------------------------------------------------------------------------
// Host launcher. Inputs: [imgs, classifications, regressions, anchors].
// ---------------------------------------------------------------------------
extern "C" void kernel_launch(void* const* d_in, const int* in_sizes, int n_in,
                              void* d_out, int out_size, void* d_ws, size_t ws_size,
                              hipStream_t stream) {
    (void)n_in; (void)out_size; (void)d_ws; (void)ws_size;

    const float* cls = (const float*)d_in[1];
    const float* reg = (const float*)d_in[2];
    const float* anc = (const float*)d_in[3];
    float* out = (float*)d_out;

    const int N = in_sizes[3] / 4;                       // anchors: [1,N,4]
    const int B = in_sizes[2] / (4 * N);                 // regressions: [B,N,4]
    const int C = (int)((long long)in_sizes[1] / ((long long)B * N)); // [B,N,C]
    const int nRows = B * N;

    const long long hw = (long long)in_sizes[0] / (3LL * B); // imgs: [B,3,H,W]
    const int S = (int)(sqrt((double)hw) + 0.5);             // H == W
    const float imgW = (float)S, imgH = (float)S;

    const int threads = 256;                              // 8 waves (wave32)
    const int wavesPerBlock = threads / 32;
    int blocks = (nRows + wavesPerBlock - 1) / wavesPerBlock;
    if (blocks > 4096) blocks = 4096;                     // grid-stride pipelining
    if (blocks < 1) blocks = 1;

    if (C == 500) {
        argmax_box_async_kernel<<<blocks, threads, 0, stream>>>(
            cls, reg, anc, out, nRows, N, imgW, imgH);
    } else {
        argmax_box_generic_kernel<<<blocks, threads, 0, stream>>>(
            cls, reg, anc, out, nRows, N, C, imgW, imgH);
    }
}